// GNN_50543175139388
// MI455X (gfx1250) — compile-verified
//
#include <hip/hip_runtime.h>
#include <hip/hip_bf16.h>

typedef float v2f __attribute__((ext_vector_type(2)));
typedef float v8f __attribute__((ext_vector_type(8)));

#define NN      100000
#define NE      1600000
#define D       128
#define NG      64
#define NLAYERS 3

// ---------------- small utility kernels ----------------

__global__ void fill_zero_kernel(float* __restrict__ p, int n) {
    int i = blockIdx.x * blockDim.x + threadIdx.x;
    if (i < n) p[i] = 0.0f;
}

__global__ void degree_kernel(const long long* __restrict__ dst,
                              float* __restrict__ deg, int E) {
    int e = blockIdx.x * blockDim.x + threadIdx.x;
    if (e < E) unsafeAtomicAdd(&deg[dst[e]], 1.0f);
}

__global__ void invdeg_kernel(float* __restrict__ deg, int n) {
    int i = blockIdx.x * blockDim.x + threadIdx.x;
    if (i < n) deg[i] = 1.0f / fmaxf(deg[i], 1.0f);
}

__global__ void batch_count_kernel(const long long* __restrict__ batch,
                                   float* __restrict__ cnt, int n) {
    int i = blockIdx.x * blockDim.x + threadIdx.x;
    if (i < n) unsafeAtomicAdd(&cnt[batch[i]], 1.0f);
}

// ---------------- WMMA GEMM: out = hin @ W^T + b ---------------------------
// 32 nodes per block (two 16-row A tiles), 8 waves -> 8 column tiles of 16.
// Each B fragment is reused by 2 (or 4, dual) WMMAs -> halves weight traffic.
// A (16x4 f32, 2 VGPR): lanes 0-15 = K=0,1 ; lanes 16-31 = K=2,3 (ISA layout).
// C/D (16x16 f32, 8 VGPR): VGPR r: lanes0-15 -> M=r, lanes16-31 -> M=r+8.

#define LDS_STRIDE 132  // 128 + 4 pad: avoids 16-way LDS bank conflicts

template<bool DUAL>
__global__ __launch_bounds__(256)
void gemm_kernel(const float* __restrict__ hin,
                 const float* __restrict__ Wl, const float* __restrict__ bl,
                 const float* __restrict__ Wr, const float* __restrict__ br,
                 float* __restrict__ outl, float* __restrict__ outr) {
    __shared__ float sA[32 * LDS_STRIDE];

    const int node0 = blockIdx.x * 32;
    const int tid   = threadIdx.x;

    // stage 32x128 activation tile into LDS (float4, padded rows)
    for (int i = tid; i < 32 * (D / 4); i += 256) {
        int row = i >> 5;          // 0..31
        int c4  = i & 31;          // 0..31 (float4 index)
        *(float4*)(&sA[row * LDS_STRIDE + c4 * 4]) =
            *(const float4*)(hin + (size_t)(node0 + row) * D + c4 * 4);
    }
    __syncthreads();

    const int wave = tid >> 5;          // 0..7 -> output column tile
    const int lane = tid & 31;
    const int o0   = wave * 16;
    const int half = lane >> 4;         // 0 or 1
    const int l16  = lane & 15;

    v8f accL0 = {};
    v8f accL1 = {};
    v8f accR0 = {};
    v8f accR1 = {};

    for (int k = 0; k < D; k += 4) {
        const int ka = k + half * 2;

        // A fragments for both 16-row tiles
        v2f a0, a1;
        a0.x = sA[l16 * LDS_STRIDE + ka];
        a0.y = sA[l16 * LDS_STRIDE + ka + 1];
        a1.x = sA[(16 + l16) * LDS_STRIDE + ka];
        a1.y = sA[(16 + l16) * LDS_STRIDE + ka + 1];

        // B fragment: B[k][n] = W[o0+n][k]; lanes 0-15: K=k,k+1, lanes 16-31: K=k+2,k+3
        const float* wl = Wl + (size_t)(o0 + l16) * D + ka;
        v2f bL; bL.x = wl[0]; bL.y = wl[1];
        accL0 = __builtin_amdgcn_wmma_f32_16x16x4_f32(false, a0, false, bL,
                                                      (short)0, accL0, false, false);
        accL1 = __builtin_amdgcn_wmma_f32_16x16x4_f32(false, a1, false, bL,
                                                      (short)0, accL1, false, false);
        if (DUAL) {
            const float* wr = Wr + (size_t)(o0 + l16) * D + ka;
            v2f bR; bR.x = wr[0]; bR.y = wr[1];
            accR0 = __builtin_amdgcn_wmma_f32_16x16x4_f32(false, a0, false, bR,
                                                          (short)0, accR0, false, false);
            accR1 = __builtin_amdgcn_wmma_f32_16x16x4_f32(false, a1, false, bR,
                                                          (short)0, accR1, false, false);
        }
    }

    const float biasL = bl[o0 + l16];
#pragma unroll
    for (int r = 0; r < 8; ++r) {
        const int row0 = node0 + r + half * 8;
        outl[(size_t)row0 * D + o0 + l16]        = accL0[r] + biasL;
        outl[(size_t)(row0 + 16) * D + o0 + l16] = accL1[r] + biasL;
    }
    if (DUAL) {
        const float biasR = br[o0 + l16];
#pragma unroll
        for (int r = 0; r < 8; ++r) {
            const int row0 = node0 + r + half * 8;
            outr[(size_t)row0 * D + o0 + l16]        = accR0[r] + biasR;
            outr[(size_t)(row0 + 16) * D + o0 + l16] = accR1[r] + biasR;
        }
    }
}

// ---------------- edge scatter: out[dst] += hr[src] * invdeg[dst] ----------
// One wave per edge; 32 lanes x float4 = 128 features. Atomics hit L2
// (hr/out working set ~100MB fits the 192MB L2).

__global__ __launch_bounds__(256)
void scatter_kernel(const long long* __restrict__ src,
                    const long long* __restrict__ dst,
                    const float* __restrict__ hr,
                    const float* __restrict__ invdeg,
                    float* __restrict__ out, int E) {
    long long idx = (long long)blockIdx.x * blockDim.x + threadIdx.x;
    if (idx >= (long long)E * 32) return;
    const int e  = (int)(idx >> 5);
    const int f4 = (int)(idx & 31) << 2;

    const long long s = src[e];
    const long long d = dst[e];
    const float wgt = invdeg[d];

    const float4 v = *(const float4*)(hr + (size_t)s * D + f4);
    float* op = out + (size_t)d * D + f4;
    unsafeAtomicAdd(op + 0, v.x * wgt);
    unsafeAtomicAdd(op + 1, v.y * wgt);
    unsafeAtomicAdd(op + 2, v.z * wgt);
    unsafeAtomicAdd(op + 3, v.w * wgt);
}

// ---------------- row L2-normalize + ReLU (one wave32 per node) ------------

__global__ __launch_bounds__(256)
void norm_relu_kernel(const float* __restrict__ out, float* __restrict__ h, int N) {
    const int wave = threadIdx.x >> 5;
    const int lane = threadIdx.x & 31;
    const int node = blockIdx.x * 8 + wave;
    if (node >= N) return;

    float4 v = *(const float4*)(out + (size_t)node * D + lane * 4);
    float ss = v.x * v.x + v.y * v.y + v.z * v.z + v.w * v.w;
#pragma unroll
    for (int off = 16; off > 0; off >>= 1) ss += __shfl_xor(ss, off, 32);

    const float inv = 1.0f / fmaxf(sqrtf(ss), 1e-12f);
    float4 r;
    r.x = fmaxf(v.x * inv, 0.0f);
    r.y = fmaxf(v.y * inv, 0.0f);
    r.z = fmaxf(v.z * inv, 0.0f);
    r.w = fmaxf(v.w * inv, 0.0f);
    *(float4*)(h + (size_t)node * D + lane * 4) = r;
}

// ---------------- final 128->16 projection + per-graph pooling -------------

__global__ __launch_bounds__(256)
void mlp_pool_kernel(const float* __restrict__ t, const long long* __restrict__ batch,
                     const float* __restrict__ W2, const float* __restrict__ b2,
                     float* __restrict__ pooled, int N) {
    const int wave = threadIdx.x >> 5;
    const int lane = threadIdx.x & 31;
    const int node = blockIdx.x * 8 + wave;
    if (node >= N) return;

    const float4 v = *(const float4*)(t + (size_t)node * D + lane * 4);
    const long long g = batch[node];

    float mine = 0.0f;
#pragma unroll
    for (int j = 0; j < 16; ++j) {
        const float4 w = *(const float4*)(W2 + (size_t)j * D + lane * 4);
        float p = v.x * w.x + v.y * w.y + v.z * w.z + v.w * w.w;
#pragma unroll
        for (int off = 16; off > 0; off >>= 1) p += __shfl_xor(p, off, 32);
        if (lane == j) mine = p + b2[j];   // xor-reduce leaves sum in all lanes
    }
    if (lane < 16) unsafeAtomicAdd(&pooled[(size_t)g * 16 + lane], mine);
}

// ---------------- mean + log_softmax over 64 graphs ------------------------

__global__ void finalize_kernel(const float* __restrict__ pooled,
                                const float* __restrict__ cnt,
                                float* __restrict__ outp) {
    const int g = threadIdx.x;
    if (g >= NG) return;
    const float c = fmaxf(cnt[g], 1.0f);
    float vals[16];
    float m = -INFINITY;
#pragma unroll
    for (int j = 0; j < 16; ++j) {
        vals[j] = pooled[g * 16 + j] / c;
        m = fmaxf(m, vals[j]);
    }
    float s = 0.0f;
#pragma unroll
    for (int j = 0; j < 16; ++j) s += expf(vals[j] - m);
    const float ls = logf(s);
#pragma unroll
    for (int j = 0; j < 16; ++j) outp[g * 16 + j] = vals[j] - m - ls;
}

// ---------------- host driver ----------------------------------------------

extern "C" void kernel_launch(void* const* d_in, const int* in_sizes, int n_in,
                              void* d_out, int out_size, void* d_ws, size_t ws_size,
                              hipStream_t stream) {
    const float*     x     = (const float*)d_in[0];
    const long long* edge  = (const long long*)d_in[1];   // [2, NE] row-major
    const long long* src   = edge;
    const long long* dstp  = edge + NE;
    const long long* batch = (const long long*)d_in[2];
    const float* Wl = (const float*)d_in[3];
    const float* bl = (const float*)d_in[4];
    const float* Wr = (const float*)d_in[5];
    const float* br = (const float*)d_in[6];
    const float* W1 = (const float*)d_in[7];
    const float* b1 = (const float*)d_in[8];
    const float* W2 = (const float*)d_in[9];
    const float* b2 = (const float*)d_in[10];
    float* outp = (float*)d_out;

    float* w = (float*)d_ws;
    const size_t nd = (size_t)NN * D;
    float* hrB    = w;            // N x 128  lin_r output
    float* outB   = w + nd;       // N x 128  hl + scattered messages / MLP temp
    float* hB     = w + 2 * nd;   // N x 128  layer activations
    float* deg    = w + 3 * nd;   // N        degree -> invdeg (in place)
    float* cnt    = deg + NN;     // 64       nodes per graph
    float* pooled = cnt + NG;     // 64 x 16

    // zero deg + cnt + pooled (contiguous)
    {
        const int zn = NN + NG + NG * 16;
        fill_zero_kernel<<<(zn + 255) / 256, 256, 0, stream>>>(deg, zn);
    }
    degree_kernel<<<(NE + 255) / 256, 256, 0, stream>>>(dstp, deg, NE);
    invdeg_kernel<<<(NN + 255) / 256, 256, 0, stream>>>(deg, NN);
    batch_count_kernel<<<(NN + 255) / 256, 256, 0, stream>>>(batch, cnt, NN);

    const float* hin = x;
    for (int i = 0; i < NLAYERS; ++i) {
        gemm_kernel<true><<<NN / 32, 256, 0, stream>>>(
            hin, Wl + (size_t)i * D * D, bl + (size_t)i * D,
            Wr + (size_t)i * D * D, br + (size_t)i * D,
            outB, hrB);
        const long long work = (long long)NE * 32;
        scatter_kernel<<<(int)((work + 255) / 256), 256, 0, stream>>>(
            src, dstp, hrB, deg, outB, NE);
        norm_relu_kernel<<<(NN + 7) / 8, 256, 0, stream>>>(outB, hB, NN);
        hin = hB;
    }

    // post-MP MLP: t = h @ W1^T + b1 (WMMA), then 128->16 + pooling
    gemm_kernel<false><<<NN / 32, 256, 0, stream>>>(hin, W1, b1,
                                                    nullptr, nullptr, outB, nullptr);
    mlp_pool_kernel<<<(NN + 7) / 8, 256, 0, stream>>>(outB, batch, W2, b2, pooled, NN);
    finalize_kernel<<<1, 64, 0, stream>>>(pooled, cnt, outp);
}